// three_gcn_25357486916245
// MI455X (gfx1250) — compile-verified
//
#include <hip/hip_runtime.h>
#include <math.h>

#define N_NODES 100000
#define N_EDGES 1600000
#define IN_SIZE 128
#define OUT_SIZE 64

typedef __attribute__((ext_vector_type(2))) float v2f;
typedef __attribute__((ext_vector_type(8))) float v8f;

// ---------------------------------------------------------------------------
// Degree accumulation: deg_src[src[e]] += 1, deg_dst[dst[e]] += 1
// ---------------------------------------------------------------------------
__global__ void degree_kernel(const int* __restrict__ src, const int* __restrict__ dst,
                              float* __restrict__ deg_src, float* __restrict__ deg_dst,
                              int n_edges) {
    int e = blockIdx.x * blockDim.x + threadIdx.x;
    if (e < n_edges) {
        atomicAdd(&deg_src[src[e]], 1.0f);
        atomicAdd(&deg_dst[dst[e]], 1.0f);
    }
}

// norm = clip(deg, 1)^-0.5, in place for both arrays
__global__ void norm_kernel(float* __restrict__ a, float* __restrict__ b, int n) {
    int i = blockIdx.x * blockDim.x + threadIdx.x;
    if (i < n) {
        a[i] = rsqrtf(fmaxf(a[i], 1.0f));
        b[i] = rsqrtf(fmaxf(b[i], 1.0f));
    }
}

// ---------------------------------------------------------------------------
// Edge-parallel SpMM scatter: agg[dst[e]] += feat[src[e]] * out_norm[src[e]]
// Each thread moves one float4 (4 features) of one edge; consecutive lanes
// cover consecutive features -> coalesced gather, coalesced atomics.
// F = 1<<LOGF features per row; LOGF-2 = log2(float4 groups per row).
// ---------------------------------------------------------------------------
template<int LOGF>
__global__ void scatter_kernel(const float* __restrict__ feat,
                               const float* __restrict__ out_norm,
                               const int* __restrict__ src, const int* __restrict__ dst,
                               float* __restrict__ agg, int n_edges) {
    const int LOGV = LOGF - 2;              // float4 groups per row
    const int FV = 1 << LOGV;
    int idx = blockIdx.x * blockDim.x + threadIdx.x;
    int e = idx >> LOGV;
    int g = idx & (FV - 1);
    if (e < n_edges) {
        int s = src[e];
        int d = dst[e];
        float sc = out_norm[s];
        const float4* frow = (const float4*)(feat + ((size_t)s << LOGF));
        float4 q = frow[g];
        float* arow = agg + ((size_t)d << LOGF) + (g << 2);
        atomicAdd(arow + 0, q.x * sc);
        atomicAdd(arow + 1, q.y * sc);
        atomicAdd(arow + 2, q.z * sc);
        atomicAdd(arow + 3, q.w * sc);
    }
}

// ---------------------------------------------------------------------------
// Dense GEMM: out[node][n] = act( (m[node][:] * in_norm[node]) @ W + bias )
// One 128-thread block = 4 waves; each wave owns one 16x16 output tile
// (node tile x 16-column tile). K consumed 4 per V_WMMA_F32_16X16X4_F32.
//
// A frag (16x4 f32): lanes 0-15 -> M=lane, VGPR{0,1}=K{k0,k0+1};
//                    lanes 16-31 -> M=lane-16, VGPR{0,1}=K{k0+2,k0+3}.
// B frag (4x16 f32): lanes 0-15 -> N=lane, VGPR{0,1}=K{k0,k0+1};
//                    lanes 16-31 -> N=lane-16, VGPR{0,1}=K{k0+2,k0+3}.
// C (16x16 f32, 8 VGPRs): VGPR v, lanes 0-15 -> M=v, N=lane;
//                                  lanes 16-31 -> M=v+8, N=lane-16.
// mode: 0 = ELU epilogue, 1 = raw logits.
// ---------------------------------------------------------------------------
__global__ void gcn_gemm_wmma_kernel(const float* __restrict__ m,
                                     const float* __restrict__ in_norm,
                                     const float* __restrict__ W,
                                     const float* __restrict__ bias,
                                     float* __restrict__ out,
                                     int K, int mode) {
    const int N = OUT_SIZE;                       // 64
    int wave = threadIdx.x >> 5;                  // 0..3 -> column tile
    int lane = threadIdx.x & 31;
    int half = lane >> 4;                         // 0 or 1
    int l16  = lane & 15;
    int node_base = blockIdx.x * 16;              // 100000 % 16 == 0
    int n_base = wave * 16;

    int arow = node_base + l16;                   // A-matrix row for this lane
    float ascale = in_norm[arow];                 // fold in-degree norm into A
    const float* mrow = m + (size_t)arow * K;
    int bcol = n_base + l16;                      // B-matrix column for this lane

    v8f c = {0.f, 0.f, 0.f, 0.f, 0.f, 0.f, 0.f, 0.f};
    for (int k0 = 0; k0 < K; k0 += 4) {
        int ka = k0 + half * 2;
        v2f a, b;
        a.x = mrow[ka]     * ascale;
        a.y = mrow[ka + 1] * ascale;
        b.x = W[(size_t)ka * N + bcol];
        b.y = W[(size_t)(ka + 1) * N + bcol];
        c = __builtin_amdgcn_wmma_f32_16x16x4_f32(
                /*neg_a=*/false, a, /*neg_b=*/false, b,
                /*c_mod=*/(short)0, c, /*reuse_a=*/false, /*reuse_b=*/false);
    }

    float bv = bias[bcol];                        // column-indexed bias
    #pragma unroll
    for (int v = 0; v < 8; ++v) {
        int row = node_base + v + half * 8;
        float val = c[v] + bv;
        if (mode == 0) val = (val > 0.0f) ? val : expm1f(val);   // ELU(alpha=1)
        out[(size_t)row * N + n_base + l16] = val;
    }
}

// ---------------------------------------------------------------------------
// Row softmax over 64 features, one 64-thread block per node.
// ---------------------------------------------------------------------------
__global__ void softmax_kernel(const float* __restrict__ logits, float* __restrict__ out) {
    __shared__ float red[64];
    int node = blockIdx.x;
    int t = threadIdx.x;
    float v = logits[(size_t)node * 64 + t];

    red[t] = v;
    __syncthreads();
    for (int s = 32; s > 0; s >>= 1) {
        if (t < s) red[t] = fmaxf(red[t], red[t + s]);
        __syncthreads();
    }
    float mx = red[0];
    __syncthreads();

    float e = expf(v - mx);
    red[t] = e;
    __syncthreads();
    for (int s = 32; s > 0; s >>= 1) {
        if (t < s) red[t] += red[t + s];
        __syncthreads();
    }
    float sum = red[0];
    out[(size_t)node * 64 + t] = e / sum;
}

// ---------------------------------------------------------------------------
extern "C" void kernel_launch(void* const* d_in, const int* in_sizes, int n_in,
                              void* d_out, int out_size, void* d_ws, size_t ws_size,
                              hipStream_t stream) {
    (void)in_sizes; (void)n_in; (void)out_size; (void)ws_size;

    const float* x  = (const float*)d_in[0];
    const float* W1 = (const float*)d_in[1];
    const float* b1 = (const float*)d_in[2];
    const float* W2 = (const float*)d_in[3];
    const float* b2 = (const float*)d_in[4];
    const float* W3 = (const float*)d_in[5];
    const float* b3 = (const float*)d_in[6];
    const int*  src = (const int*)d_in[7];
    const int*  dst = (const int*)d_in[8];

    float* out = (float*)d_out;
    float* h1 = out;                                  // [N,64]
    float* h2 = out + (size_t)N_NODES * OUT_SIZE;     // [N,64]
    float* h3 = out + (size_t)2 * N_NODES * OUT_SIZE; // [N,64]

    float* ws       = (float*)d_ws;
    float* out_norm = ws;                             // [N]
    float* in_norm  = ws + N_NODES;                   // [N]
    float* agg      = ws + 2 * (size_t)N_NODES;       // [N,128] accumulation buffer
    float* logits   = agg + (size_t)N_NODES * 64;     // layer-3 logits [N,64] reuse the second half of agg's floats

    const int TB = 256;

    // --- degree norms -------------------------------------------------------
    hipMemsetAsync(out_norm, 0, 2 * (size_t)N_NODES * sizeof(float), stream);
    degree_kernel<<<(N_EDGES + TB - 1) / TB, TB, 0, stream>>>(src, dst, out_norm, in_norm, N_EDGES);
    norm_kernel<<<(N_NODES + TB - 1) / TB, TB, 0, stream>>>(out_norm, in_norm, N_NODES);

    // --- layer 1: x (128) -> h1 (64), ELU ----------------------------------
    hipMemsetAsync(agg, 0, (size_t)N_NODES * IN_SIZE * sizeof(float), stream);
    {
        long long work = (long long)N_EDGES * (IN_SIZE / 4);
        scatter_kernel<7><<<(unsigned)((work + TB - 1) / TB), TB, 0, stream>>>(
            x, out_norm, src, dst, agg, N_EDGES);
    }
    gcn_gemm_wmma_kernel<<<N_NODES / 16, 128, 0, stream>>>(
        agg, in_norm, W1, b1, h1, IN_SIZE, /*mode=*/0);

    // --- layer 2: h1 (64) -> h2 (64), ELU ----------------------------------
    hipMemsetAsync(agg, 0, (size_t)N_NODES * OUT_SIZE * sizeof(float), stream);
    {
        long long work = (long long)N_EDGES * (OUT_SIZE / 4);
        scatter_kernel<6><<<(unsigned)((work + TB - 1) / TB), TB, 0, stream>>>(
            h1, out_norm, src, dst, agg, N_EDGES);
    }
    gcn_gemm_wmma_kernel<<<N_NODES / 16, 128, 0, stream>>>(
        agg, in_norm, W2, b2, h2, OUT_SIZE, /*mode=*/0);

    // --- layer 3: h2 (64) -> softmax(h3) -----------------------------------
    hipMemsetAsync(agg, 0, (size_t)N_NODES * OUT_SIZE * sizeof(float), stream);
    {
        long long work = (long long)N_EDGES * (OUT_SIZE / 4);
        scatter_kernel<6><<<(unsigned)((work + TB - 1) / TB), TB, 0, stream>>>(
            h2, out_norm, src, dst, agg, N_EDGES);
    }
    gcn_gemm_wmma_kernel<<<N_NODES / 16, 128, 0, stream>>>(
        agg, in_norm, W3, b3, logits, OUT_SIZE, /*mode=*/1);
    softmax_kernel<<<N_NODES, 64, 0, stream>>>(logits, h3);
}